// QuantizedRetriever_80616536145980
// MI455X (gfx1250) — compile-verified
//
#include <hip/hip_runtime.h>
#include <hip/hip_bf16.h>
#include <math.h>

// Problem constants (match reference)
#define TFRAMES 8192
#define DIM     256
#define PVOC    64
#define KCENT   128

#define MAXTILES 572            // worst case: 8192 + 64*15 padded frames = 9152 -> /16
#define PADT     9152

typedef __attribute__((ext_vector_type(2))) float v2f;
typedef __attribute__((ext_vector_type(8))) float v8f;

// ---------------- workspace layout (ints) ----------------
// [0..63]      counts
// [64..127]    cursor
// [128..192]   padded offsets (65)
// [256..827]   tile_phone (MAXTILES)
// [1024..10175] order (PADT)
// [10240..]    norm2 (8192 floats)
#define WS_COUNTS 0
#define WS_CURSOR 64
#define WS_POFF   128
#define WS_TPHONE 256
#define WS_ORDER  1024
#define WS_NORM2  10240

// ---- kernel 1: init scratch ----
__global__ void qr_init(int* __restrict__ wsi) {
    int idx = blockIdx.x * blockDim.x + threadIdx.x;
    if (idx < PVOC) { wsi[WS_COUNTS + idx] = 0; wsi[WS_CURSOR + idx] = 0; }
    if (idx < MAXTILES) wsi[WS_TPHONE + idx] = -1;
    if (idx < PADT)     wsi[WS_ORDER + idx]  = -1;
}

// ---- kernel 2: per-phone histogram ----
__global__ void qr_count(const int* __restrict__ phones, int* __restrict__ wsi) {
    int t = blockIdx.x * blockDim.x + threadIdx.x;
    if (t >= TFRAMES) return;
    int p = phones[t];
    atomicAdd(&wsi[WS_COUNTS + p], 1);
}

// ---- kernel 3: padded prefix sum + tile->phone map (tiny serial) ----
__global__ void qr_offsets(int* __restrict__ wsi) {
    if (threadIdx.x != 0 || blockIdx.x != 0) return;
    int off = 0;
    for (int p = 0; p < PVOC; ++p) {
        wsi[WS_POFF + p] = off;
        int padded = ((wsi[WS_COUNTS + p] + 15) / 16) * 16;
        int t0 = off / 16, t1 = (off + padded) / 16;
        for (int t = t0; t < t1; ++t) wsi[WS_TPHONE + t] = p;
        off += padded;
    }
    wsi[WS_POFF + PVOC] = off;
}

// ---- kernel 4: scatter frame ids into padded buckets ----
__global__ void qr_scatter(const int* __restrict__ phones, int* __restrict__ wsi) {
    int t = blockIdx.x * blockDim.x + threadIdx.x;
    if (t >= TFRAMES) return;
    int p = phones[t];
    int pos = atomicAdd(&wsi[WS_CURSOR + p], 1);
    wsi[WS_ORDER + wsi[WS_POFF + p] + pos] = t;
}

// ---- kernel 5: squared norms, one wave per center ----
__global__ void qr_norms(const float* __restrict__ centers, float* __restrict__ norm2) {
    int wavesPerBlock = blockDim.x / 32;
    int c = blockIdx.x * wavesPerBlock + (threadIdx.x / 32);   // 0..P*K-1
    int lane = threadIdx.x & 31;
    const float* row = centers + (size_t)c * DIM;
    float s = 0.f;
    #pragma unroll
    for (int i = lane; i < DIM; i += 32) { float v = row[i]; s += v * v; }
    #pragma unroll
    for (int off = 16; off > 0; off >>= 1) s += __shfl_xor(s, off, 32);
    if (lane == 0) norm2[c] = s;
}

// ---- kernel 6: WMMA distance + argmin + gather, one block per 16-frame tile ----
// Each wave owns 32 centers = two 16-wide N-tiles, computed in ONE fused k-loop:
// a single LDS A-load feeds two independent WMMA accumulator chains.
__launch_bounds__(128)
__global__ void qr_main(const float* __restrict__ h,
                        const float* __restrict__ centers,
                        const int* __restrict__ wsi,
                        const float* __restrict__ norm2,
                        float* __restrict__ out) {
    int tile = blockIdx.x;
    int p = wsi[WS_TPHONE + tile];
    if (p < 0) return;                       // block-uniform: EXEC stays all-1s below

    __shared__ int   rows[16];
    __shared__ float hs[16][DIM + 1];        // +1 pad: kill 64-bank conflicts
    __shared__ float scores[16][KCENT + 1];
    __shared__ int   best[16];

    int tid = threadIdx.x;
    if (tid < 16) rows[tid] = wsi[WS_ORDER + tile * 16 + tid];
    __syncthreads();

    // stage 16 gathered h rows into LDS (invalid rows -> 0)
    for (int i = tid; i < 16 * DIM; i += 128) {
        int r = i / DIM, c = i % DIM;
        int t = rows[r];
        hs[r][c] = (t >= 0) ? h[(size_t)t * DIM + c] : 0.f;
    }
    __syncthreads();

    int wave = tid >> 5;
    int lane = tid & 31;
    int m = lane & 15;                       // A row / B column within tile
    int khalf = (lane < 16) ? 0 : 2;         // K sub-pair held by this lane half
    const float* Cp  = centers + (size_t)p * KCENT * DIM;
    const float* nrm = norm2 + (size_t)p * KCENT;

    int n_lo = wave * 32 + m;                // column in N-tile 0
    int n_hi = n_lo + 16;                    // column in N-tile 1
    const float* crow0 = Cp + (size_t)n_lo * DIM + khalf;
    const float* crow1 = Cp + (size_t)n_hi * DIM + khalf;
    const float* arow  = &hs[m][khalf];

    v8f acc0 = {};
    v8f acc1 = {};
    #pragma unroll 4
    for (int k0 = 0; k0 < DIM; k0 += 4) {
        v2f a, b0, b1;
        a[0]  = arow[k0];     a[1]  = arow[k0 + 1];
        b0[0] = crow0[k0];    b0[1] = crow0[k0 + 1];
        b1[0] = crow1[k0];    b1[1] = crow1[k0 + 1];
        acc0 = __builtin_amdgcn_wmma_f32_16x16x4_f32(
            false, a, false, b0, (short)0, acc0, false, false);
        acc1 = __builtin_amdgcn_wmma_f32_16x16x4_f32(
            false, a, false, b1, (short)0, acc1, false, false);
    }

    // lane holds rows {rbase..rbase+7} at columns n_lo and n_hi
    int rbase = (lane < 16) ? 0 : 8;
    float nn0 = nrm[n_lo];
    float nn1 = nrm[n_hi];
    #pragma unroll
    for (int r = 0; r < 8; ++r) {
        scores[rbase + r][n_lo] = nn0 - 2.0f * acc0[r];
        scores[rbase + r][n_hi] = nn1 - 2.0f * acc1[r];
    }
    __syncthreads();

    // per-row argmin over 128 centers; '<' keeps first min like jnp.argmin
    if (tid < 16) {
        float bv = INFINITY; int bi = 0;
        for (int k = 0; k < KCENT; ++k) {
            float v = scores[tid][k];
            if (v < bv) { bv = v; bi = k; }
        }
        best[tid] = bi;
    }
    __syncthreads();

    // emit winning center rows (coalesced per row)
    for (int i = tid; i < 16 * DIM; i += 128) {
        int r = i / DIM, c = i % DIM;
        int t = rows[r];
        if (t >= 0)
            out[(size_t)t * DIM + c] = Cp[(size_t)best[r] * DIM + c];
    }
}

extern "C" void kernel_launch(void* const* d_in, const int* in_sizes, int n_in,
                              void* d_out, int out_size, void* d_ws, size_t ws_size,
                              hipStream_t stream) {
    const float* h       = (const float*)d_in[0];
    const float* centers = (const float*)d_in[1];
    const int*   phones  = (const int*)d_in[2];
    float* out = (float*)d_out;
    int*   wsi = (int*)d_ws;
    float* norm2 = (float*)(wsi + WS_NORM2);

    qr_init<<<(PADT + 255) / 256, 256, 0, stream>>>(wsi);
    qr_count<<<TFRAMES / 256, 256, 0, stream>>>(phones, wsi);
    qr_offsets<<<1, 32, 0, stream>>>(wsi);
    qr_scatter<<<TFRAMES / 256, 256, 0, stream>>>(phones, wsi);
    qr_norms<<<(PVOC * KCENT) / 4, 128, 0, stream>>>(centers, norm2);
    qr_main<<<MAXTILES, 128, 0, stream>>>(h, centers, wsi, norm2, out);
}